// UniFcosNet_52055003628262
// MI455X (gfx1250) — compile-verified
//
#include <hip/hip_runtime.h>

typedef __attribute__((ext_vector_type(2))) float v2f;
typedef __attribute__((ext_vector_type(8))) float v8f;

#define NUM_POINT 9
#define B_DIM 2
#define C_DIM 256
#define H_DIM 100
#define W_DIM 100
#define HW (H_DIM * W_DIM)            // 10000
#define CHW (C_DIM * HW)              // 2,560,000
#define N_LOC HW                      // 10000 locations
#define TOTAL_SAMPLES (B_DIM * N_LOC * NUM_POINT)   // 180000
#define SAMPLES_PER_WAVE 16
#define WAVES_TOTAL (TOTAL_SAMPLES / SAMPLES_PER_WAVE) // 11250 (exact)

// ---------------------------------------------------------------------------
// deform_feats: each wave32 produces 16 samples x 256 channels (16 KB) using
// v_wmma_f32_16x16x4_f32 with a block-diagonal weight matrix A:
//   D[s, c] = sum_s A_s x B_s,  A_s[m,k] = (m==s) ? w_k(s) : 0,
//   B_s[k, c] = bilinear corner k of sample s, channel c.
// This is exact f32 bilinear interpolation routed through the XDL unit.
// ---------------------------------------------------------------------------
__global__ __launch_bounds__(256) void deform_feats_wmma(
    const float* __restrict__ feat,       // (B, C, H, W)
    const float* __restrict__ locations,  // (N, 3)
    const float* __restrict__ offset,     // (B, N, P, 2)
    float* __restrict__ out)              // (B, N, P, C)
{
    const int lane = threadIdx.x & 31;
    const int wave = blockIdx.x * (blockDim.x >> 5) + (threadIdx.x >> 5);
    if (wave >= WAVES_TOTAL) return;          // wave-uniform: EXEC stays all-1

    const int m  = lane & 15;   // matrix row slot / B-column (channel-in-block)
    const int hi = lane >> 4;   // half-wave: selects K pair {0,1} vs {2,3}
    const int sample_base = wave * SAMPLES_PER_WAVE;

    // ---- per-sample parameters, computed in lane slot m (dup in both halves)
    const int sample = sample_base + m;
    const int b   = sample / (N_LOC * NUM_POINT);
    const int rem = sample - b * (N_LOC * NUM_POINT);
    const int n   = rem / NUM_POINT;

    const float cx = locations[n * 3 + 0];
    const float cy = locations[n * 3 + 1];
    const float ox = offset[sample * 2 + 0];
    const float oy = offset[sample * 2 + 1];

    // index = (center + offset*scale_*0.1) / stride, scale_ = 8*8 = 64
    const float w = (cx + (ox * 64.0f) * 0.1f) / 8.0f;
    const float h = (cy + (oy * 64.0f) * 0.1f) / 8.0f;

    // reference bilinear_gather clamp semantics (incl. extrapolation < 0)
    const float h_low  = fminf(fmaxf(floorf(h), 0.0f), (float)(H_DIM - 1));
    const bool  h_edge = (h_low >= (float)(H_DIM - 1));
    const float h_high = h_edge ? h_low : h_low + 1.0f;
    const float h_eff  = h_edge ? h_low : h;
    const float w_low  = fminf(fmaxf(floorf(w), 0.0f), (float)(W_DIM - 1));
    const bool  w_edge = (w_low >= (float)(W_DIM - 1));
    const float w_high = w_edge ? w_low : w_low + 1.0f;
    const float w_eff  = w_edge ? w_low : w;

    const int hl = (int)h_low, wl = (int)w_low;
    const int hh = (int)h_high, wh = (int)w_high;
    const float lh  = h_eff - h_low, lw  = w_eff - w_low;
    const float hh_ = 1.0f - lh,     hw_ = 1.0f - lw;

    const float wt0 = hh_ * hw_;   // corner K0: (hl, wl)
    const float wt1 = hh_ * lw;    // corner K1: (hl, wh)
    const float wt2 = lh  * hw_;   // corner K2: (hh, wl)
    const float wt3 = lh  * lw;    // corner K3: (hh, wh)

    const int p00 = b * CHW + hl * W_DIM + wl;   // element index, channel 0
    const int dyd = (hh - hl) * W_DIM;           // row step to h_high
    const int dxd = (wh - wl);                   // col step to w_high

    // ---- broadcast each sample's scalars to the whole wave; build per-lane
    //      gather addresses (B-matrix layout) and block-diagonal A values.
    int   addrA[16], addrB[16];
    float a0v[16], a1v[16];
#pragma unroll
    for (int s = 0; s < 16; ++s) {
        const int   p00s = __shfl(p00, s, 32);
        const int   dyds = __shfl(dyd, s, 32);
        const int   dxds = __shfl(dxd, s, 32);
        const float t0   = __shfl(wt0, s, 32);
        const float t1   = __shfl(wt1, s, 32);
        const float t2   = __shfl(wt2, s, 32);
        const float t3   = __shfl(wt3, s, 32);
        // B 4x16 layout: vgpr0 -> K = 2*hi (x = w_low), vgpr1 -> K = 2*hi+1 (x = w_high)
        const int base = p00s + hi * dyds + m * HW;
        addrA[s] = base;
        addrB[s] = base + dxds;
        // A 16x4 layout: lanes 0-15 hold K0/K1 of row m, lanes 16-31 hold K2/K3
        a0v[s] = (m == s) ? (hi ? t2 : t0) : 0.0f;
        a1v[s] = (m == s) ? (hi ? t3 : t1) : 0.0f;
    }

    const int outBase = sample_base * C_DIM;

#pragma unroll 1
    for (int blk = 0; blk < 16; ++blk) {        // 16 channel blocks of 16
        const int blkoff = blk * 16 * HW;
        v8f acc = {0.f, 0.f, 0.f, 0.f, 0.f, 0.f, 0.f, 0.f};
#pragma unroll
        for (int s = 0; s < 16; ++s) {
            v2f bmat;
            bmat.x = feat[addrA[s] + blkoff];
            bmat.y = feat[addrB[s] + blkoff];
            v2f amat;
            amat.x = a0v[s];
            amat.y = a1v[s];
            acc = __builtin_amdgcn_wmma_f32_16x16x4_f32(
                false, amat, false, bmat, (short)0, acc, false, false);
        }
        // D layout: vgpr r / lane -> row M = r + 8*hi (sample slot), col = m (channel)
#pragma unroll
        for (int r = 0; r < 8; ++r) {
            const int oidx = outBase + (r + 8 * hi) * C_DIM + blk * 16 + m;
            out[oidx] = acc[r];
        }
    }
}

// ---------------------------------------------------------------------------
// boxes: one thread per (b, n); 9-point min/max, float4 store.
// ---------------------------------------------------------------------------
__global__ __launch_bounds__(256) void regression_boxes_kernel(
    const float* __restrict__ locations,  // (N, 3)
    const float* __restrict__ offset,     // (B, N, P, 2)
    const float* __restrict__ loc_pred,   // (B, N, P*2)
    float* __restrict__ boxes)            // (B, N, 4)
{
    const int t = blockIdx.x * blockDim.x + threadIdx.x;
    if (t >= B_DIM * N_LOC) return;
    const int b = t / N_LOC;
    const int n = t - b * N_LOC;

    const float cx    = locations[n * 3 + 0];
    const float cy    = locations[n * 3 + 1];
    const float scale = locations[n * 3 + 2] * 8.0f;  // BASE_SCALE

    const float* off = offset   + ((b * N_LOC + n) * NUM_POINT) * 2;
    const float* lp  = loc_pred + ((b * N_LOC + n) * NUM_POINT) * 2;

    float xmin =  INFINITY, ymin =  INFINITY;
    float xmax = -INFINITY, ymax = -INFINITY;
#pragma unroll
    for (int p = 0; p < NUM_POINT; ++p) {
        const float px = (cx + (off[p * 2 + 0] * scale) * 0.1f) + (lp[p * 2 + 0] * scale) * 0.5f;
        const float py = (cy + (off[p * 2 + 1] * scale) * 0.1f) + (lp[p * 2 + 1] * scale) * 0.5f;
        xmin = fminf(xmin, px); xmax = fmaxf(xmax, px);
        ymin = fminf(ymin, py); ymax = fmaxf(ymax, py);
    }
    float4 r;
    r.x = xmin; r.y = ymin; r.z = xmax; r.w = ymax;
    ((float4*)boxes)[t] = r;
}

extern "C" void kernel_launch(void* const* d_in, const int* in_sizes, int n_in,
                              void* d_out, int out_size, void* d_ws, size_t ws_size,
                              hipStream_t stream) {
    (void)in_sizes; (void)n_in; (void)out_size; (void)d_ws; (void)ws_size;
    const float* feat      = (const float*)d_in[0];   // (2,256,100,100)
    const float* locations = (const float*)d_in[1];   // (10000,3)
    const float* offset    = (const float*)d_in[2];   // (2,10000,9,2)
    const float* loc_pred  = (const float*)d_in[3];   // (2,10000,18)

    float* out_feats = (float*)d_out;                              // 46,080,000 f32
    float* out_boxes = out_feats + (size_t)TOTAL_SAMPLES * C_DIM;  // 80,000 f32

    // 8 waves / block, 11250 waves total -> 1407 blocks (trailing waves exit)
    const int nblocks = (WAVES_TOTAL + 7) / 8;
    deform_feats_wmma<<<nblocks, 256, 0, stream>>>(feat, locations, offset, out_feats);

    const int nthreads = B_DIM * N_LOC;
    regression_boxes_kernel<<<(nthreads + 255) / 256, 256, 0, stream>>>(
        locations, offset, loc_pred, out_boxes);
}